// BiologicalPopulationVectorDecoder_87935160418581
// MI455X (gfx1250) — compile-verified
//
#include <hip/hip_runtime.h>

#define THREADS 256
#define MAX_BLOCKS 2048

typedef float v2f __attribute__((ext_vector_type(2)));
typedef float v4f __attribute__((ext_vector_type(4)));
typedef float v8f __attribute__((ext_vector_type(8)));

// ---------------------------------------------------------------------------
// Full-wave (32-lane) sum using V_WMMA_F32_16X16X4_F32.
//
// A is 16x4 fp32: lane m (0..15) holds row m's K=0 (vgpr0) and K=1 (vgpr1);
// lane m+16 holds row m's K=2 (vgpr0) and K=3 (vgpr1)  [ISA 7.12.2].
// With A.x = v, A.y = 0 and B = all-ones (layout-invariant), the MAC array
// computes D[m,n] = v_m + v_{m+16} for every n. Summing the 8 D vgprs in a
// lane yields the half-wave partial (rows 0..7 for lanes<16, rows 8..15 for
// lanes>=16); one more identical WMMA folds the halves, leaving the full
// 32-lane sum broadcast to every lane. EXEC must be all ones at the call site
// (it is: called after loop reconvergence from fully-populated 256T blocks).
// ---------------------------------------------------------------------------
__device__ __forceinline__ float wave_sum32(float v) {
#if __has_builtin(__builtin_amdgcn_wmma_f32_16x16x4_f32)
    v2f a; a[0] = v;    a[1] = 0.0f;
    v2f b; b[0] = 1.0f; b[1] = 1.0f;
    v8f c = {};
    v8f d = __builtin_amdgcn_wmma_f32_16x16x4_f32(
        false, a, false, b, (short)0, c, false, false);
    float s = ((d[0] + d[1]) + (d[2] + d[3])) + ((d[4] + d[5]) + (d[6] + d[7]));
    v2f a2; a2[0] = s; a2[1] = 0.0f;
    v8f d2 = __builtin_amdgcn_wmma_f32_16x16x4_f32(
        false, a2, false, b, (short)0, c, false, false);
    return d2[0];
#else
    for (int off = 16; off > 0; off >>= 1) v += __shfl_xor(v, off, 32);
    return v;
#endif
}

// Non-temporal streaming loads: 470 MB single-pass > 192 MB L2, zero reuse.
__device__ __forceinline__ float ldnt (const float* p) { return __builtin_nontemporal_load(p); }
__device__ __forceinline__ v4f   ldnt4(const v4f*   p) { return __builtin_nontemporal_load(p); }

// ---------------------------------------------------------------------------
// Pass 1: grid-stride streaming reduction. 8 fp32 accumulators per thread,
// WMMA wave reduction, fixed-order LDS combine across the 8 waves, one
// 8-float partial per block (deterministic; no float atomics).
// ---------------------------------------------------------------------------
__global__ __launch_bounds__(THREADS)
void pass1_reduce(const float* __restrict__ na, const v4f* __restrict__ w4,
                  const float* __restrict__ pd, const float* __restrict__ tw,
                  float* __restrict__ partial, int n) {
    const float HALF_PI = 1.5707963267948966f;

    float p0 = 0.f, p1 = 0.f, p2 = 0.f, p3 = 0.f;   // pop-vector dots
    float t0 = 0.f, t1 = 0.f, t2 = 0.f, t3 = 0.f;   // tuning dots

    const int stride = gridDim.x * blockDim.x;
    #pragma unroll 2
    for (int i = blockIdx.x * blockDim.x + threadIdx.x; i < n; i += stride) {
        float a = ldnt(na + i);
        a = (a > 0.001f) ? a : 0.0f;                 // spike gate

        v4f w = ldnt4(w4 + i);
        p0 = fmaf(a, w.x, p0);
        p1 = fmaf(a, w.y, p1);
        p2 = fmaf(a, w.z, p2);
        p3 = fmaf(a, w.w, p3);

        float inv  = 1.0f / ldnt(tw + i);
        float base = -ldnt(pd + i) * inv;            // (theta_0 - pd)/w
        float step = HALF_PI * inv;                  // action angle spacing / w
        t0 = fmaf(a, __cosf(base),               t0);
        t1 = fmaf(a, __cosf(base +        step), t1);
        t2 = fmaf(a, __cosf(base + 2.0f * step), t2);
        t3 = fmaf(a, __cosf(base + 3.0f * step), t3);
    }

    // 32-lane sums via the matrix pipe (all threads reconverged -> EXEC full)
    p0 = wave_sum32(p0); p1 = wave_sum32(p1);
    p2 = wave_sum32(p2); p3 = wave_sum32(p3);
    t0 = wave_sum32(t0); t1 = wave_sum32(t1);
    t2 = wave_sum32(t2); t3 = wave_sum32(t3);

    __shared__ float sm[(THREADS / 32) * 8];
    const int lane = threadIdx.x & 31;
    const int wv   = threadIdx.x >> 5;
    if (lane == 0) {
        sm[wv * 8 + 0] = p0; sm[wv * 8 + 1] = p1;
        sm[wv * 8 + 2] = p2; sm[wv * 8 + 3] = p3;
        sm[wv * 8 + 4] = t0; sm[wv * 8 + 5] = t1;
        sm[wv * 8 + 6] = t2; sm[wv * 8 + 7] = t3;
    }
    __syncthreads();
    if (threadIdx.x < 8) {                           // fixed-order wave combine
        float s = 0.0f;
        #pragma unroll
        for (int w_ = 0; w_ < THREADS / 32; ++w_) s += sm[w_ * 8 + threadIdx.x];
        partial[blockIdx.x * 8 + threadIdx.x] = s;
    }
}

// ---------------------------------------------------------------------------
// Pass 2: single block. Deterministic fixed-grouping reduction of the block
// partials, then the O(1) tail: combined, competition, both softmaxes.
// ---------------------------------------------------------------------------
__global__ __launch_bounds__(THREADS)
void pass2_finalize(const float* __restrict__ partial, int nblocks,
                    const float* __restrict__ cw, const float* __restrict__ inh,
                    float* __restrict__ out) {
    __shared__ float red[32 * 8];
    const int t = threadIdx.x;
    const int c = t & 7;        // component 0..7
    const int j = t >> 3;       // slot 0..31

    float s = 0.0f;
    for (int g = j; g < nblocks; g += 32) s += partial[g * 8 + c];
    red[j * 8 + c] = s;
    __syncthreads();
    for (int st = 16; st >= 1; st >>= 1) {
        if (j < st) red[j * 8 + c] += red[(j + st) * 8 + c];
        __syncthreads();
    }

    if (t == 0) {
        float pop[4], tun[4], comb[4], comp[4];
        #pragma unroll
        for (int a = 0; a < 4; ++a) {
            pop[a]  = red[a];
            tun[a]  = red[4 + a];
            comb[a] = 2.0f * pop[a] + 0.5f * tun[a];   // POP_VECTOR_GAIN, 0.5
        }
        const float ih = inh[0];
        #pragma unroll
        for (int a = 0; a < 4; ++a) {
            float dot = 0.0f;
            #pragma unroll
            for (int b = 0; b < 4; ++b) dot = fmaf(cw[a * 4 + b], comb[b], dot);
            comp[a] = comb[a] - ih * dot;
        }
        // softmax(combined)
        float m1 = fmaxf(fmaxf(comb[0], comb[1]), fmaxf(comb[2], comb[3]));
        float e1[4], z1 = 0.0f;
        #pragma unroll
        for (int a = 0; a < 4; ++a) { e1[a] = __expf(comb[a] - m1); z1 += e1[a]; }
        // softmax(competitive * WTA_GAIN)
        float w3[4];
        #pragma unroll
        for (int a = 0; a < 4; ++a) w3[a] = comp[a] * 3.0f;
        float m2 = fmaxf(fmaxf(w3[0], w3[1]), fmaxf(w3[2], w3[3]));
        float e2[4], z2 = 0.0f;
        #pragma unroll
        for (int a = 0; a < 4; ++a) { e2[a] = __expf(w3[a] - m2); z2 += e2[a]; }

        #pragma unroll
        for (int a = 0; a < 4; ++a) {
            out[0  + a] = e1[a] / z1;   // action_probabilities
            out[4  + a] = e2[a] / z2;   // enhanced_probabilities
            out[8  + a] = comp[a];      // competitive
            out[12 + a] = pop[a];       // action_activations
            out[16 + a] = tun[a];       // tuning_contributions
        }
    }
}

extern "C" void kernel_launch(void* const* d_in, const int* in_sizes, int n_in,
                              void* d_out, int out_size, void* d_ws, size_t ws_size,
                              hipStream_t stream) {
    const float* na  = (const float*)d_in[0];
    const v4f*   w4  = (const v4f*)  d_in[1];
    const float* pd  = (const float*)d_in[2];
    const float* tw  = (const float*)d_in[3];
    const float* cw  = (const float*)d_in[4];
    const float* inh = (const float*)d_in[5];
    float* out     = (float*)d_out;
    float* partial = (float*)d_ws;
    const int n = in_sizes[0];

    int G = MAX_BLOCKS;                                  // 2048 * 8 * 4B = 64 KB ws
    while ((size_t)G * 8 * sizeof(float) > ws_size && G > 1) G >>= 1;

    hipLaunchKernelGGL(pass1_reduce, dim3(G), dim3(THREADS), 0, stream,
                       na, w4, pd, tw, partial, n);
    hipLaunchKernelGGL(pass2_finalize, dim3(1), dim3(THREADS), 0, stream,
                       partial, G, cw, inh, out);
    (void)n_in; (void)out_size;
}